// CosSimRouter_pad_49778670960794
// MI455X (gfx1250) — compile-verified
//
#include <hip/hip_runtime.h>
#include <math.h>

// ---------------------------------------------------------------------------
// CosSimRouter for MI455X (gfx1250, wave32, WMMA f32 16x16x4)
//
//  K1 norms_kernel   : row L2 norms of vision(576x4096) & text(8192x4096)
//  K2 scores_kernel  : scores[i] = max_j cos(v_i, t_j) -- fp32 WMMA, 16x128/wave,
//                      software-pipelined (register double buffering)
//  K3 select_kernel  : softmax/argsort/cumsum/threshold/3x3-neighbors/unique
//  K4 smat_kernel    : S[a][i] = dot(v_uniq[a], v_i)   -- fp32 WMMA, 16x64/wave
//  K5 out_kernel     : stable top-16, softmax, weighted gather-sum
// ---------------------------------------------------------------------------

typedef __attribute__((ext_vector_type(2))) float v2f;
typedef __attribute__((ext_vector_type(8))) float v8f;

#define KD   4096
#define NV   576
#define NT   8192
#define GRID 24

// Order-preserving float <-> uint encoding so atomicMax(uint) == float max.
__device__ __forceinline__ unsigned enc_f(float x) {
  unsigned u = __float_as_uint(x);
  return (u & 0x80000000u) ? ~u : (u | 0x80000000u);
}
__device__ __forceinline__ float dec_f(unsigned e) {
  unsigned u = (e & 0x80000000u) ? (e & 0x7FFFFFFFu) : ~e;
  return __uint_as_float(u);
}

// ---------------------------------------------------------------------------
// K1: row norms (one block per row), also zero-init encoded score maxima.
// ---------------------------------------------------------------------------
__global__ __launch_bounds__(256) void norms_kernel(
    const float* __restrict__ v, const float* __restrict__ t,
    float* __restrict__ vnorm, float* __restrict__ tnorm,
    unsigned int* __restrict__ sbits) {
  __shared__ float part[8];
  int b = blockIdx.x;
  const float* row;
  float* dst;
  if (b < NV) { row = v + (size_t)b * KD; dst = vnorm + b; }
  else        { row = t + (size_t)(b - NV) * KD; dst = tnorm + (b - NV); }
  float sum = 0.f;
  for (int k = threadIdx.x; k < KD; k += 256) {
    float x = row[k];
    sum = fmaf(x, x, sum);
  }
  #pragma unroll
  for (int d = 1; d < 32; d <<= 1) sum += __shfl_xor(sum, d, 32);
  if ((threadIdx.x & 31) == 0) part[threadIdx.x >> 5] = sum;
  __syncthreads();
  if (threadIdx.x == 0) {
    float tot = 0.f;
    #pragma unroll
    for (int w = 0; w < 8; ++w) tot += part[w];
    *dst = fmaxf(sqrtf(tot), 1e-8f);
    if (b < NV) sbits[b] = 0u;  // below enc of any finite float
  }
}

// ---------------------------------------------------------------------------
// K2: scores[i] = max_j (mask[j] ? dot(v_i,t_j)/(|v_i||t_j|) : 0)
// Wave tile: 16 rows x 128 cols, 8 independent f32 accumulators.
// Register double-buffered K loop: loads for step k+4 issue before the
// WMMAs consuming step k, so loads stay in flight across matrix ops.
// A frag: m=lane&15 -> vision row, koff=(lane>>4)*2 -> contiguous float2.
// B frag: n=lane&15 -> text row,  same koff         -> contiguous float2.
// ---------------------------------------------------------------------------
__global__ __launch_bounds__(256) void scores_kernel(
    const float* __restrict__ v, const float* __restrict__ t,
    const unsigned char* __restrict__ mask,
    const float* __restrict__ vnorm, const float* __restrict__ tnorm,
    unsigned int* __restrict__ sbits) {
  const int wave = threadIdx.x >> 5;
  const int lane = threadIdx.x & 31;
  const int rowBase = blockIdx.x * 16;                   // 36 row tiles
  const int colBase = (blockIdx.y * 8 + wave) * 128;     // 8*8 waves cover 8192
  const int m = lane & 15;
  const int half = lane >> 4;
  const int koff = half * 2;

  const float* ap = v + (size_t)(rowBase + m) * KD + koff;
  const float* bp[8];
  #pragma unroll
  for (int j = 0; j < 8; ++j)
    bp[j] = t + (size_t)(colBase + j * 16 + m) * KD + koff;

  v8f c[8];
  #pragma unroll
  for (int j = 0; j < 8; ++j) c[j] = (v8f){};

  // Prologue: load step k=0 fragments.
  v2f aC = *(const v2f*)(ap);
  v2f bC[8];
  #pragma unroll
  for (int j = 0; j < 8; ++j) bC[j] = *(const v2f*)(bp[j]);

  #pragma unroll 2
  for (int k = 4; k < KD; k += 4) {
    // Issue next-step loads first so they overlap the WMMAs below.
    v2f aN = *(const v2f*)(ap + k);
    v2f bN[8];
    #pragma unroll
    for (int j = 0; j < 8; ++j) bN[j] = *(const v2f*)(bp[j] + k);
    #pragma unroll
    for (int j = 0; j < 8; ++j)
      c[j] = __builtin_amdgcn_wmma_f32_16x16x4_f32(false, aC, false, bC[j],
                                                   (short)0, c[j], false, false);
    aC = aN;
    #pragma unroll
    for (int j = 0; j < 8; ++j) bC[j] = bN[j];
  }
  // Epilogue: consume the last fragments.
  #pragma unroll
  for (int j = 0; j < 8; ++j)
    c[j] = __builtin_amdgcn_wmma_f32_16x16x4_f32(false, aC, false, bC[j],
                                                 (short)0, c[j], false, false);

  // C layout: vgpr r, lanes<16 -> M=r, lanes>=16 -> M=r+8; N = lane&15.
  float it[8];
  #pragma unroll
  for (int j = 0; j < 8; ++j) {
    const int col = colBase + j * 16 + m;
    it[j] = mask[col] ? (1.0f / tnorm[col]) : 0.0f;  // mask==0 -> value 0.0
  }

  #pragma unroll
  for (int r = 0; r < 8; ++r) {
    const int M = rowBase + r + 8 * half;
    const float iv = 1.0f / vnorm[M];
    float best = c[0][r] * iv * it[0];
    #pragma unroll
    for (int j = 1; j < 8; ++j) best = fmaxf(best, c[j][r] * iv * it[j]);
    #pragma unroll
    for (int d = 1; d < 16; d <<= 1) best = fmaxf(best, __shfl_xor(best, d, 16));
    if (m == 0) atomicMax(&sbits[M], enc_f(best));
  }
}

// ---------------------------------------------------------------------------
// K3: softmax(scores/0.05), stable descending argsort, sequential cumsum,
// thr = #(cum <= 0.5), mark clipped 3x3 neighbors of selected, compact unique.
// Single block; sequential parts done by thread 0 to match the reference's
// ordering semantics exactly and stay deterministic across graph replays.
// ---------------------------------------------------------------------------
__global__ __launch_bounds__(576) void select_kernel(
    const unsigned int* __restrict__ sbits,
    int* __restrict__ uniq, int* __restrict__ Uptr) {
  __shared__ float probs[NV];
  __shared__ float psort[NV];
  __shared__ int order[NV];
  __shared__ int flags[NV];
  __shared__ float red[2];
  const int i = threadIdx.x;

  float x = dec_f(sbits[i]) * 20.0f;  // scores / TEMP, TEMP = 0.05
  probs[i] = x;
  flags[i] = 0;
  __syncthreads();
  if (i == 0) {
    float mx = probs[0];
    for (int j = 1; j < NV; ++j) mx = fmaxf(mx, probs[j]);
    red[0] = mx;
  }
  __syncthreads();
  probs[i] = expf(x - red[0]);
  __syncthreads();
  if (i == 0) {
    float sum = 0.f;
    for (int j = 0; j < NV; ++j) sum += probs[j];
    red[1] = sum;
  }
  __syncthreads();
  const float p = probs[i] / red[1];
  probs[i] = p;
  __syncthreads();
  int rank = 0;
  for (int j = 0; j < NV; ++j) {
    float pj = probs[j];
    rank += (pj > p) || (pj == p && j < i);  // stable descending
  }
  order[rank] = i;
  psort[rank] = p;
  __syncthreads();
  if (i == 0) {
    float cum = 0.f;
    int thr = 0;
    for (int k = 0; k < NV; ++k) { cum += psort[k]; if (cum <= 0.5f) thr++; }
    for (int k = 0; k < thr; ++k) {
      int sel = order[k];
      int rr = sel / GRID, cc = sel % GRID;
      #pragma unroll
      for (int di = -1; di <= 1; ++di)
        #pragma unroll
        for (int dj = -1; dj <= 1; ++dj) {
          if (di == 0 && dj == 0) continue;
          int r2 = min(max(rr + di, 0), GRID - 1);
          int c2 = min(max(cc + dj, 0), GRID - 1);
          flags[r2 * GRID + c2] = 1;
        }
    }
    int U = 0;
    for (int j = 0; j < NV; ++j) if (flags[j]) uniq[U++] = j;
    *Uptr = U;
    for (int j = U; j < NV; ++j) uniq[j] = 0;  // padded rows for the WMMA tiles
  }
}

// ---------------------------------------------------------------------------
// K4: S[a][i] = dot(v_uniq[a], v_i) via fp32 WMMA; 16x64 tile per wave,
// register double-buffered like K2.
// ---------------------------------------------------------------------------
__global__ __launch_bounds__(32) void smat_kernel(
    const float* __restrict__ v, const int* __restrict__ uniq,
    const int* __restrict__ Uptr, float* __restrict__ S) {
  const int U = *Uptr;
  const int rowBase = blockIdx.x * 16;
  if (rowBase >= U) return;
  const int lane = threadIdx.x & 31;
  const int m = lane & 15;
  const int half = lane >> 4;
  const int koff = half * 2;
  const int colBase = blockIdx.y * 64;
  const int urow = uniq[rowBase + m];

  const float* ap = v + (size_t)urow * KD + koff;
  const float* bp[4];
  #pragma unroll
  for (int j = 0; j < 4; ++j)
    bp[j] = v + (size_t)(colBase + j * 16 + m) * KD + koff;

  v8f c[4];
  #pragma unroll
  for (int j = 0; j < 4; ++j) c[j] = (v8f){};

  v2f aC = *(const v2f*)(ap);
  v2f bC[4];
  #pragma unroll
  for (int j = 0; j < 4; ++j) bC[j] = *(const v2f*)(bp[j]);

  #pragma unroll 2
  for (int k = 4; k < KD; k += 4) {
    v2f aN = *(const v2f*)(ap + k);
    v2f bN[4];
    #pragma unroll
    for (int j = 0; j < 4; ++j) bN[j] = *(const v2f*)(bp[j] + k);
    #pragma unroll
    for (int j = 0; j < 4; ++j)
      c[j] = __builtin_amdgcn_wmma_f32_16x16x4_f32(false, aC, false, bC[j],
                                                   (short)0, c[j], false, false);
    aC = aN;
    #pragma unroll
    for (int j = 0; j < 4; ++j) bC[j] = bN[j];
  }
  #pragma unroll
  for (int j = 0; j < 4; ++j)
    c[j] = __builtin_amdgcn_wmma_f32_16x16x4_f32(false, aC, false, bC[j],
                                                 (short)0, c[j], false, false);

  #pragma unroll
  for (int r = 0; r < 8; ++r) {
    const int a = rowBase + r + 8 * half;
    #pragma unroll
    for (int j = 0; j < 4; ++j)
      S[(size_t)a * NV + colBase + j * 16 + m] = c[j][r];
  }
}

// ---------------------------------------------------------------------------
// K5: per output row: cosine-scale S row, stable top-16 (lax.top_k semantics),
// softmax over 16, weighted gather-sum of vision rows -> out.
// ---------------------------------------------------------------------------
__global__ __launch_bounds__(256) void out_kernel(
    const float* __restrict__ v, const float* __restrict__ vnorm,
    const int* __restrict__ uniq, const int* __restrict__ Uptr,
    const float* __restrict__ S, float* __restrict__ out) {
  __shared__ float vals[NV];
  __shared__ float tv[16];
  __shared__ int ti[16];
  __shared__ float tw[16];
  const int U = *Uptr;
  const int b = blockIdx.x;
  if (b >= U) return;
  const int r = uniq[b];
  const float ivr = 1.0f / vnorm[r];

  for (int i = threadIdx.x; i < NV; i += blockDim.x)
    vals[i] = S[(size_t)b * NV + i] * ivr / vnorm[i];
  __syncthreads();

  for (int i = threadIdx.x; i < NV; i += blockDim.x) {
    float vi = vals[i];
    int rank = 0;
    for (int j = 0; j < NV; ++j) {
      float vj = vals[j];
      rank += (vj > vi) || (vj == vi && j < i);
    }
    if (rank < 16) { ti[rank] = i; tv[rank] = vi; }
  }
  __syncthreads();

  if (threadIdx.x == 0) {
    float mx = tv[0];  // sorted descending
    float e[16];
    float sum = 0.f;
    #pragma unroll
    for (int k = 0; k < 16; ++k) { e[k] = expf(tv[k] - mx); sum += e[k]; }
    #pragma unroll
    for (int k = 0; k < 16; ++k) tw[k] = e[k] / sum;
  }
  __syncthreads();

  for (int d = threadIdx.x; d < KD; d += blockDim.x) {
    float acc = 0.f;
    #pragma unroll
    for (int k = 0; k < 16; ++k)
      acc = fmaf(tw[k], v[(size_t)ti[k] * KD + d], acc);
    out[(size_t)b * KD + d] = acc;
  }
}

// ---------------------------------------------------------------------------
extern "C" void kernel_launch(void* const* d_in, const int* in_sizes, int n_in,
                              void* d_out, int out_size, void* d_ws, size_t ws_size,
                              hipStream_t stream) {
  const float* v = (const float*)d_in[0];               // 576 x 4096
  const float* t = (const float*)d_in[1];               // 8192 x 4096
  const unsigned char* mask = (const unsigned char*)d_in[2];  // 8192 bool
  float* out = (float*)d_out;

  float* ws = (float*)d_ws;
  float* vnorm = ws;                                    // 576
  float* tnorm = ws + 576;                              // 8192
  unsigned int* sbits = (unsigned int*)(ws + 8768);     // 576
  int* uniq = (int*)(ws + 9344);                        // 576
  int* Uptr = (int*)(ws + 9920);                        // 1
  float* S = ws + 10240;                                // 576 x 576

  norms_kernel<<<NV + NT, 256, 0, stream>>>(v, t, vnorm, tnorm, sbits);
  scores_kernel<<<dim3(NV / 16, 8), 256, 0, stream>>>(v, t, mask, vnorm, tnorm, sbits);
  select_kernel<<<1, NV, 0, stream>>>(sbits, uniq, Uptr);
  smat_kernel<<<dim3(NV / 16, NV / 64), 32, 0, stream>>>(v, uniq, Uptr, S);
  out_kernel<<<NV, 256, 0, stream>>>(v, vnorm, uniq, Uptr, S, out);
}